// GCN_test_13881334301058
// MI455X (gfx1250) — compile-verified
//
#include <hip/hip_runtime.h>
#include <hip/hip_bf16.h>

typedef __attribute__((ext_vector_type(2))) float v2f;
typedef __attribute__((ext_vector_type(8))) float v8f;

#define N0 1408
#define EE 90112
#define NPAD 2048

// ---------------- edge init ----------------
__global__ void init_edges_k(const int* __restrict__ ei, int* __restrict__ row,
                             int* __restrict__ col, float* __restrict__ ew, int E) {
    int e = blockIdx.x * blockDim.x + threadIdx.x;
    if (e >= E) return;
    row[e] = ei[e];
    col[e] = ei[E + e];
    ew[e]  = 1.0f;
}

// ---------------- dense GEMM: H[n x 16] = A[n x K] @ W[K x 16], fp32 WMMA ----------------
// One wave (32 lanes) per 16-row tile. Exact fp32 via V_WMMA_F32_16X16X4_F32.
__global__ void gemm16_wmma_k(const float* __restrict__ A, const float* __restrict__ W,
                              float* __restrict__ H, int K) {
    const int lane = threadIdx.x;       // 0..31, full EXEC required for WMMA
    const int half = lane >> 4;         // 0 or 1
    const int m    = lane & 15;
    const int row  = blockIdx.x * 16 + m;

    v8f c = {};
    for (int kk = 0; kk < K; kk += 4) {
        const int k0 = kk + 2 * half;   // A: lanes0-15 hold K pair {kk,kk+1}; lanes16-31 {kk+2,kk+3}
        v2f a, b;
        a.x = A[row * K + k0];
        a.y = A[row * K + k0 + 1];
        b.x = W[k0 * 16 + m];           // B 4x16: same K-pair split across lane halves
        b.y = W[(k0 + 1) * 16 + m];
        c = __builtin_amdgcn_wmma_f32_16x16x4_f32(
                /*neg_a=*/false, a, /*neg_b=*/false, b,
                /*c_mod=*/(short)0, c, /*reuse_a=*/false, /*reuse_b=*/false);
    }
    // C/D layout: VGPR j -> M = j + 8*half, N = lane&15
    const int outbase = (blockIdx.x * 16 + half * 8) * 16 + m;
#pragma unroll
    for (int j = 0; j < 8; ++j)
        H[outbase + j * 16] = c[j];
}

// ---------------- degree / norm ----------------
__global__ void deg_init_k(float* __restrict__ deg, int n) {
    int i = blockIdx.x * blockDim.x + threadIdx.x;
    if (i < n) deg[i] = 1.0f;  // self-loop weight
}

__global__ void deg_scatter_k(const int* __restrict__ col, const float* __restrict__ ew,
                              float* __restrict__ deg, int E) {
    int e = blockIdx.x * blockDim.x + threadIdx.x;
    if (e >= E) return;
    float w = ew[e];
    if (w > 0.0f) unsafeAtomicAdd(&deg[col[e]], w);
}

__global__ void dinv_k(const float* __restrict__ deg, float* __restrict__ dinv, int n) {
    int i = blockIdx.x * blockDim.x + threadIdx.x;
    if (i < n) dinv[i] = rsqrtf(deg[i]);   // deg >= 1 always (self-loop)
}

// out[i,f] = dinv[i]^2 * h[i,f]   (self-loop contribution; also initializes accumulator)
__global__ void agg_init_k(const float* __restrict__ h, const float* __restrict__ dinv,
                           float* __restrict__ out, int total) {
    int i = blockIdx.x * blockDim.x + threadIdx.x;
    if (i >= total) return;
    float d = dinv[i >> 4];
    out[i] = d * d * h[i];
}

__global__ void agg_scatter_k(const int* __restrict__ row, const int* __restrict__ col,
                              const float* __restrict__ ew, const float* __restrict__ dinv,
                              const float* __restrict__ h, float* __restrict__ out, int E) {
    int e = blockIdx.x * blockDim.x + threadIdx.x;
    if (e >= E) return;
    float w = ew[e];
    if (w <= 0.0f) return;
    int r = row[e], c = col[e];
    float nm = dinv[r] * w * dinv[c];
    const float* hr = h + r * 16;
    float* oc = out + c * 16;
#pragma unroll
    for (int f = 0; f < 16; ++f)
        unsafeAtomicAdd(&oc[f], nm * hr[f]);
}

__global__ void bias_relu_k(const float* __restrict__ in, const float* __restrict__ b,
                            float* __restrict__ out, int total) {
    int i = blockIdx.x * blockDim.x + threadIdx.x;
    if (i >= total) return;
    out[i] = fmaxf(in[i] + b[i & 15], 0.0f);
}

// ---------------- pooling ----------------
__global__ void score_k(const float* __restrict__ x, const float* __restrict__ pw,
                        float* __restrict__ sc, int n) {
    int i = blockIdx.x * blockDim.x + threadIdx.x;
    if (i >= n) return;
    float nrm = 0.0f, d = 0.0f;
#pragma unroll
    for (int f = 0; f < 16; ++f) nrm += pw[f] * pw[f];
    nrm = sqrtf(nrm);
#pragma unroll
    for (int f = 0; f < 16; ++f) d += x[i * 16 + f] * pw[f];
    sc[i] = tanhf(d / nrm);
}

// One workgroup: bitonic sort of (score desc, idx asc) over NPAD padded slots,
// then emit pos[] and pooled features x[perm[i]] * val[i].
__global__ void topk_k(const float* __restrict__ sc, const float* __restrict__ xin,
                       float* __restrict__ xout, int* __restrict__ pos, int n, int k) {
    __shared__ float skey[NPAD];
    __shared__ int   sidx[NPAD];
    const int t = threadIdx.x;  // 0..1023
    for (int i = t; i < NPAD; i += 1024) {
        skey[i] = (i < n) ? sc[i] : -__builtin_inff();
        sidx[i] = (i < n) ? i : 0x7fffffff;
    }
    __syncthreads();
    for (int size = 2; size <= NPAD; size <<= 1) {
        for (int stride = size >> 1; stride > 0; stride >>= 1) {
            int a = ((t & ~(stride - 1)) << 1) | (t & (stride - 1));
            int b = a | stride;
            bool up = ((a & size) == 0);
            float ka = skey[a], kb = skey[b];
            int   ia = sidx[a], ib = sidx[b];
            bool a_before_b = (ka > kb) || (ka == kb && ia < ib);  // desired order
            bool sw = up ? !a_before_b : a_before_b;
            if (sw) { skey[a] = kb; skey[b] = ka; sidx[a] = ib; sidx[b] = ia; }
            __syncthreads();
        }
    }
    for (int i = t; i < n; i += 1024) pos[i] = -1;
    __syncthreads();
    for (int i = t; i < k; i += 1024) pos[sidx[i]] = i;
    for (int i = t; i < k * 16; i += 1024) {
        int node = i >> 4, f = i & 15;
        xout[i] = xin[sidx[node] * 16 + f] * skey[node];
    }
}

__global__ void remap_edges_k(int* __restrict__ row, int* __restrict__ col,
                              float* __restrict__ ew, const int* __restrict__ pos, int E) {
    int e = blockIdx.x * blockDim.x + threadIdx.x;
    if (e >= E) return;
    int nr = pos[row[e]];
    int nc = pos[col[e]];
    float w = ew[e];
    bool valid = (nr >= 0) && (nc >= 0) && (w > 0.0f);
    ew[e]  = valid ? w : 0.0f;
    row[e] = valid ? nr : 0;
    col[e] = valid ? nc : 0;
}

// ---------------- final MLP (GEMV, bandwidth-bound) ----------------
__global__ void gemv_k(const float* __restrict__ h, const float* __restrict__ W,
                       const float* __restrict__ b, float* __restrict__ out, int K, int N) {
    int j = blockIdx.x * blockDim.x + threadIdx.x;
    if (j >= N) return;
    float acc = b[j];
    for (int i = 0; i < K; ++i)
        acc = fmaf(h[i], W[i * N + j], acc);
    out[j] = acc;
}

extern "C" void kernel_launch(void* const* d_in, const int* in_sizes, int n_in,
                              void* d_out, int out_size, void* d_ws, size_t ws_size,
                              hipStream_t stream) {
    (void)in_sizes; (void)n_in; (void)out_size; (void)ws_size;

    const float* x_in       = (const float*)d_in[0];
    const int*   edge_index = (const int*)d_in[1];
    const float* Wl[4]  = {(const float*)d_in[3], (const float*)d_in[5],
                           (const float*)d_in[7], (const float*)d_in[9]};
    const float* Bl[4]  = {(const float*)d_in[4], (const float*)d_in[6],
                           (const float*)d_in[8], (const float*)d_in[10]};
    const float* PWl[4] = {(const float*)d_in[11], (const float*)d_in[12],
                           (const float*)d_in[13], (const float*)d_in[14]};
    const float* fw1 = (const float*)d_in[15];
    const float* fb1 = (const float*)d_in[16];
    const float* fw2 = (const float*)d_in[17];
    const float* fb2 = (const float*)d_in[18];
    const float* fw3 = (const float*)d_in[19];
    const float* fb3 = (const float*)d_in[20];

    // workspace carving (256B aligned)
    char* base = (char*)d_ws;
    size_t off = 0;
    auto carve = [&](size_t bytes) -> void* {
        void* r = base + off;
        off += (bytes + 255) & ~(size_t)255;
        return r;
    };
    int*   row_w = (int*)carve(EE * 4);
    int*   col_w = (int*)carve(EE * 4);
    float* ew_w  = (float*)carve(EE * 4);
    float* deg   = (float*)carve(N0 * 4);
    float* dinv  = (float*)carve(N0 * 4);
    float* hbuf  = (float*)carve(N0 * 16 * 4);
    float* obuf  = (float*)carve(N0 * 16 * 4);
    float* xr    = (float*)carve(N0 * 16 * 4);
    float* pb0   = (float*)carve(N0 * 16 * 4);
    float* pb1   = (float*)carve(N0 * 16 * 4);
    float* sc    = (float*)carve(N0 * 4);
    int*   pos   = (int*)carve(N0 * 4);
    float* fc1b  = (float*)carve(1024 * 4);
    float* fc2b  = (float*)carve(512 * 4);

    const int TB = 256;
    const int EG = (EE + TB - 1) / TB;

    init_edges_k<<<EG, TB, 0, stream>>>(edge_index, row_w, col_w, ew_w, EE);

    int n = N0;
    int K = 128;
    const float* gin = x_in;
    for (int L = 0; L < 4; ++L) {
        gemm16_wmma_k<<<n / 16, 32, 0, stream>>>(gin, Wl[L], hbuf, K);
        deg_init_k<<<(n + TB - 1) / TB, TB, 0, stream>>>(deg, n);
        deg_scatter_k<<<EG, TB, 0, stream>>>(col_w, ew_w, deg, EE);
        dinv_k<<<(n + TB - 1) / TB, TB, 0, stream>>>(deg, dinv, n);
        agg_init_k<<<(n * 16 + TB - 1) / TB, TB, 0, stream>>>(hbuf, dinv, obuf, n * 16);
        agg_scatter_k<<<EG, TB, 0, stream>>>(row_w, col_w, ew_w, dinv, hbuf, obuf, EE);
        bias_relu_k<<<(n * 16 + TB - 1) / TB, TB, 0, stream>>>(obuf, Bl[L], xr, n * 16);
        score_k<<<(n + TB - 1) / TB, TB, 0, stream>>>(xr, PWl[L], sc, n);
        int k = n / 2;  // n is even every round: ceil(n/2) == n/2
        float* pooled = (L & 1) ? pb1 : pb0;
        topk_k<<<1, 1024, 0, stream>>>(sc, xr, pooled, pos, n, k);
        remap_edges_k<<<EG, TB, 0, stream>>>(row_w, col_w, ew_w, pos, EE);
        gin = pooled;
        K = 16;
        n = k;
    }

    // gin now holds the 88*16 = 1408 flattened features
    gemv_k<<<(1024 + TB - 1) / TB, TB, 0, stream>>>(gin, fw1, fb1, fc1b, 1408, 1024);
    gemv_k<<<(512 + TB - 1) / TB, TB, 0, stream>>>(fc1b, fw2, fb2, fc2b, 1024, 512);
    gemv_k<<<(96 + TB - 1) / TB, TB, 0, stream>>>(fc2b, fw3, fb3, (float*)d_out, 512, 96);
}